// RVQA_56538949484663
// MI455X (gfx1250) — compile-verified
//
#include <hip/hip_runtime.h>
#include <math.h>

// VQ-VAE fused forward for MI455X (gfx1250, wave32, WMMA f32 16x16x4 + TDM).
// N=131072, IN_DIM=256, D=64, K=1024.

typedef float v2f __attribute__((ext_vector_type(2)));
typedef float v8f __attribute__((ext_vector_type(8)));
typedef unsigned int v4u __attribute__((ext_vector_type(4)));
typedef int v4i __attribute__((ext_vector_type(4)));
typedef int v8i __attribute__((ext_vector_type(8)));

#define NROWS  131072
#define INDIM  256
#define DDIM   64
#define KCODES 1024

#define TILE_M 64          // rows per workgroup
#define XS_STRIDE 260      // 256 + 4 pad: row-strided lane reads hit 16 distinct banks
#define ZS_STRIDE 68       // 64 + 4 pad

// LDS byte offset of a __shared__ object: AS(3) pointer value == LDS offset.
__device__ static inline unsigned int lds_offset_bytes(const void* p) {
    return (unsigned int)(unsigned long long)
        (__attribute__((address_space(3))) const void*)p;
}

// --------------------------------------------------------------------------
// Pre-pass: ||c_k||^2 for all codes (tiny; stream-ordered before main kernel)
// --------------------------------------------------------------------------
__global__ void RVQA_cnorm_kernel(const float* __restrict__ cb,
                                  float* __restrict__ cn) {
    int k = blockIdx.x * blockDim.x + threadIdx.x;
    if (k < KCODES) {
        const float* row = cb + (size_t)k * DDIM;
        float s = 0.0f;
        #pragma unroll 8
        for (int d = 0; d < DDIM; ++d) { float v = row[d]; s += v * v; }
        cn[k] = s;
    }
}

// --------------------------------------------------------------------------
// Fused encoder -> VQ argmin -> gather -> decoder. 256 threads = 8 wave32.
// --------------------------------------------------------------------------
__launch_bounds__(256)
__global__ void RVQA_fused_kernel(const float* __restrict__ x,
                                  const float* __restrict__ W_enc,
                                  const float* __restrict__ b_enc,
                                  const float* __restrict__ codebook,
                                  const float* __restrict__ W_dec,
                                  const float* __restrict__ b_dec,
                                  const float* __restrict__ cnorm,
                                  float* __restrict__ recon,
                                  float* __restrict__ zq_out,
                                  int* __restrict__ zind_out) {
    __shared__ float xs[TILE_M * XS_STRIDE];   // x tile,   64x256 (TDM row-padded)
    __shared__ float zs[TILE_M * ZS_STRIDE];   // z_e then z_q, 64x64 (padded)
    __shared__ float redv[2 * TILE_M];         // per-half best distance
    __shared__ int   redi[2 * TILE_M];         // per-half best code
    __shared__ int   kfin[TILE_M];             // final code per row

    const int tid  = threadIdx.x;
    const int wave = tid >> 5;                 // 0..7
    const int lane = tid & 31;
    const int l16  = lane & 15;                // col/row position within 16
    const int hi   = lane >> 4;                // half-wave select (K / M split)
    const int row0 = blockIdx.x * TILE_M;

    // ---- Stage 0: DMA x tile into padded LDS via the Tensor Data Mover ----
#if __has_builtin(__builtin_amdgcn_tensor_load_to_lds)
    if (wave == 0) {
        unsigned long long ga =
            (unsigned long long)(const void*)(x + (size_t)row0 * INDIM);
        unsigned int lds = lds_offset_bytes(xs);

        // D# Group 0: count=1 | lds_addr | global_addr[56:0] | type=2
        v4u g0;
        g0.x = 1u;                                   // count=1, user descriptor
        g0.y = lds;                                  // lds_addr (bytes)
        g0.z = (unsigned int)(ga & 0xFFFFFFFFull);   // global_addr[31:0]
        g0.w = (unsigned int)((ga >> 32) & 0x01FFFFFFull) | 0x80000000u; // +type=2

        // D# Group 1: data_size=4B(enc 2), pad_enable, pad_interval=256 DW
        // (enc 7), pad_amount=4 DW (enc 3); tensor 256x64, tile 256x64,
        // dim0_stride=256 elements.  (256 floats data + 4 floats pad = 260.)
        v8i g1;
        g1[0] = (2 << 16) | (1 << 20) | (7 << 22) | (3 << 25);
        g1[1] = (INDIM & 0xFFFF) << 16;              // tensor_dim0[15:0]
        g1[2] = (TILE_M & 0xFFFF) << 16;             // dim0[31:16]=0 | tensor_dim1[15:0]
        g1[3] = (INDIM & 0xFFFF) << 16;              // dim1[31:16]=0 | tile_dim0
        g1[4] = (TILE_M & 0xFFFF);                   // tile_dim1 | tile_dim2=0
        g1[5] = INDIM;                               // tensor_dim0_stride[31:0]
        g1[6] = 0;                                   // stride[47:32] | dim1_stride lo
        g1[7] = 0;

        v4i g2 = {0, 0, 0, 0};                       // 2D tensor: groups 2/3 unused
        v4i g3 = {0, 0, 0, 0};
        v8i g4 = {0, 0, 0, 0, 0, 0, 0, 0};           // extra group (6-arg toolchain)
        __builtin_amdgcn_tensor_load_to_lds(g0, g1, g2, g3, g4, 0);
        __builtin_amdgcn_s_wait_tensorcnt(0);
    }
#else
    {
        const float4* xg = reinterpret_cast<const float4*>(x + (size_t)row0 * INDIM);
        #pragma unroll
        for (int i = 0; i < 16; ++i) {
            int e  = tid + i * 256;
            int r  = e >> 6, c4 = e & 63;
            float4 v = xg[r * (INDIM / 4) + c4];
            float* dst = &xs[r * XS_STRIDE + c4 * 4];
            dst[0] = v.x; dst[1] = v.y; dst[2] = v.z; dst[3] = v.w;
        }
    }
#endif
    __syncthreads();

    // ---- Stage 1: encoder z_e = tanh(x @ W_enc + b_enc) ----------------
    // 16 output tiles (4 mi x 4 ni); each wave computes 2.
    #pragma unroll
    for (int tt = 0; tt < 2; ++tt) {
        int t  = wave * 2 + tt;
        int mi = t >> 2, ni = t & 3;
        int arow = mi * 16 + l16;              // A: M = lane position
        int col  = ni * 16 + l16;              // B/D: N = lane position
        v8f acc = {};
        for (int kb = 0; kb < INDIM; kb += 4) {
            int k0 = kb + hi * 2;              // A layout: lanes 16-31 hold K+2,K+3
            v2f a, b;
            a.x = xs[arow * XS_STRIDE + k0];
            a.y = xs[arow * XS_STRIDE + k0 + 1];
            b.x = W_enc[(size_t)k0 * DDIM + col];
            b.y = W_enc[(size_t)(k0 + 1) * DDIM + col];
            acc = __builtin_amdgcn_wmma_f32_16x16x4_f32(
                      false, a, false, b, (short)0, acc, false, false);
        }
        float bias = b_enc[col];
        #pragma unroll
        for (int r = 0; r < 8; ++r) {          // D layout: VGPR r -> row 8*hi + r
            float v = tanhf(acc[r] + bias);
            zs[(mi * 16 + hi * 8 + r) * ZS_STRIDE + col] = v;
        }
    }
    __syncthreads();

    // ---- Stage 2: VQ — argmin_k (||c_k||^2 - 2 z.c_k) ------------------
    // Row-tile mi = wave>>1; each of the two waves scans 512 codes.
    {
        int mi   = wave >> 1;
        int half = wave & 1;
        int arow = mi * 16 + l16;

        v2f afrag[16];                          // z_e A-fragments, reused x32
        #pragma unroll
        for (int d16 = 0; d16 < 16; ++d16) {
            int d0 = d16 * 4 + hi * 2;
            afrag[d16].x = zs[arow * ZS_STRIDE + d0];
            afrag[d16].y = zs[arow * ZS_STRIDE + d0 + 1];
        }

        float best[8]; int bidx[8];
        #pragma unroll
        for (int r = 0; r < 8; ++r) { best[r] = 3.4e38f; bidx[r] = 0; }

        for (int ci = 0; ci < 32; ++ci) {
            int code = half * 512 + ci * 16 + l16;      // B column = code id
            const float* cb = codebook + (size_t)code * DDIM;  // B[d][code] = cb[d]
            v8f acc = {};
            #pragma unroll
            for (int d16 = 0; d16 < 16; ++d16) {
                int d0 = d16 * 4 + hi * 2;
                v2f b; b.x = cb[d0]; b.y = cb[d0 + 1];
                acc = __builtin_amdgcn_wmma_f32_16x16x4_f32(
                          false, afrag[d16], false, b, (short)0, acc, false, false);
            }
            float cn = cnorm[code];
            #pragma unroll
            for (int r = 0; r < 8; ++r) {
                float s = cn - 2.0f * acc[r];
                if (s < best[r]) { best[r] = s; bidx[r] = code; }  // ascending per lane
            }
        }
        // Reduce across the 16 lanes of each half-wave (cols of the tile).
        #pragma unroll
        for (int m = 1; m < 16; m <<= 1) {
            #pragma unroll
            for (int r = 0; r < 8; ++r) {
                float ov = __shfl_xor(best[r], m, 32);
                int   oi = __shfl_xor(bidx[r], m, 32);
                if (ov < best[r] || (ov == best[r] && oi < bidx[r])) {
                    best[r] = ov; bidx[r] = oi;
                }
            }
        }
        if (l16 == 0) {                         // lane 0 -> rows mi*16+r; lane 16 -> +8
            #pragma unroll
            for (int r = 0; r < 8; ++r) {
                int row = mi * 16 + hi * 8 + r;
                redv[half * TILE_M + row] = best[r];
                redi[half * TILE_M + row] = bidx[r];
            }
        }
    }
    __syncthreads();

    // ---- Combine the two code halves, emit z_ind -----------------------
    if (tid < TILE_M) {
        float v0 = redv[tid], v1 = redv[TILE_M + tid];
        int   i0 = redi[tid], i1 = redi[TILE_M + tid];
        int k = (v1 < v0 || (v1 == v0 && i1 < i0)) ? i1 : i0;
        kfin[tid] = k;
        zind_out[row0 + tid] = k;
    }
    __syncthreads();

    // ---- Gather z_q rows into LDS (overwrites z_e) + emit z_q ----------
    for (int i = tid; i < TILE_M * DDIM; i += 256) {
        int r = i >> 6, c = i & 63;
        float v = codebook[(size_t)kfin[r] * DDIM + c];
        zs[r * ZS_STRIDE + c] = v;
        __builtin_nontemporal_store(v, &zq_out[(size_t)(row0 + r) * DDIM + c]);
    }
    __syncthreads();

    // ---- Stage 3: decoder recon = z_q @ W_dec + b_dec ------------------
    // 64 output tiles (4 mi x 16 ni); wave -> mi = wave>>1, ni = (wave&1)*8 + j.
    {
        int mi   = wave >> 1;
        int arow = mi * 16 + l16;
        v2f afrag[16];
        #pragma unroll
        for (int d16 = 0; d16 < 16; ++d16) {
            int d0 = d16 * 4 + hi * 2;
            afrag[d16].x = zs[arow * ZS_STRIDE + d0];
            afrag[d16].y = zs[arow * ZS_STRIDE + d0 + 1];
        }
        #pragma unroll
        for (int j = 0; j < 8; ++j) {
            int ni  = (wave & 1) * 8 + j;
            int col = ni * 16 + l16;
            v8f acc = {};
            #pragma unroll
            for (int d16 = 0; d16 < 16; ++d16) {
                int d0 = d16 * 4 + hi * 2;
                v2f b;
                b.x = W_dec[(size_t)d0 * INDIM + col];
                b.y = W_dec[(size_t)(d0 + 1) * INDIM + col];
                acc = __builtin_amdgcn_wmma_f32_16x16x4_f32(
                          false, afrag[d16], false, b, (short)0, acc, false, false);
            }
            float bias = b_dec[col];
            #pragma unroll
            for (int r = 0; r < 8; ++r) {
                __builtin_nontemporal_store(
                    acc[r] + bias,
                    &recon[(size_t)(row0 + mi * 16 + hi * 8 + r) * INDIM + col]);
            }
        }
    }
}

// --------------------------------------------------------------------------
extern "C" void kernel_launch(void* const* d_in, const int* in_sizes, int n_in,
                              void* d_out, int out_size, void* d_ws, size_t ws_size,
                              hipStream_t stream) {
    const float* x        = (const float*)d_in[0];
    const float* W_enc    = (const float*)d_in[1];
    const float* b_enc    = (const float*)d_in[2];
    const float* codebook = (const float*)d_in[3];
    const float* W_dec    = (const float*)d_in[4];
    const float* b_dec    = (const float*)d_in[5];

    float* out   = (float*)d_out;
    float* recon = out;                                         // N*INDIM
    float* zq    = out + (size_t)NROWS * INDIM;                 // N*DDIM
    int*   zind  = (int*)(out + (size_t)NROWS * INDIM + (size_t)NROWS * DDIM); // N

    float* cnorm = (float*)d_ws;                                // KCODES floats

    RVQA_cnorm_kernel<<<KCODES / 256, 256, 0, stream>>>(codebook, cnorm);
    RVQA_fused_kernel<<<NROWS / TILE_M, 256, 0, stream>>>(
        x, W_enc, b_enc, codebook, W_dec, b_dec, cnorm, recon, zq, zind);
}